// RGCN_39410619908628
// MI455X (gfx1250) — compile-verified
//
#include <hip/hip_runtime.h>
#include <hip/hip_bf16.h>

typedef float v2f __attribute__((ext_vector_type(2)));
typedef float v8f __attribute__((ext_vector_type(8)));

#define R_REL 4
#define NB 2
#define DIM 128   // IN == OUT == 128

// ---------------------------------------------------------------------------
// Stage 1: w_rel[r,i,o] = comp[r,0]*weight[0,i,o] + comp[r,1]*weight[1,i,o]
// ---------------------------------------------------------------------------
__global__ void rgcn_make_wrel(const float* __restrict__ comp,
                               const float* __restrict__ weight,
                               float* __restrict__ wrel) {
    int idx = blockIdx.x * 256 + threadIdx.x;        // 0..65535
    int r   = idx >> 14;                             // / (128*128)
    int io  = idx & 16383;
    wrel[idx] = comp[r * NB + 0] * weight[io] +
                comp[r * NB + 1] * weight[DIM * DIM + io];
}

// ---------------------------------------------------------------------------
// Stage 3: SpMM scatter: out[r, rows[e], :] += vals[e] * seq[cols[e], :]
// One wave per edge; lane handles 4 consecutive floats (float4 gather burst).
// unsafeAtomicAdd -> GLOBAL_ATOMIC_ADD_F32 (no-return, STOREcnt-tracked).
// ---------------------------------------------------------------------------
__global__ __launch_bounds__(256) void rgcn_spmm(const float* __restrict__ seq,
                                                 const float* __restrict__ vals,
                                                 const int* __restrict__ rows,
                                                 const int* __restrict__ cols,
                                                 float* __restrict__ out,
                                                 int E, int N) {
    int lane = threadIdx.x & 31;
    int wave = (blockIdx.x * blockDim.x + threadIdx.x) >> 5;  // edge index
    int r = blockIdx.y;
    if (wave >= E) return;

    long long eidx = (long long)r * E + wave;
    int row = rows[eidx];
    int col = cols[eidx];
    float v = vals[eidx];

    const float4* s = (const float4*)(seq + (size_t)col * DIM);
    float4 x = s[lane];

    float* o = out + ((size_t)r * N + row) * DIM + lane * 4;
    unsafeAtomicAdd(o + 0, v * x.x);
    unsafeAtomicAdd(o + 1, v * x.y);
    unsafeAtomicAdd(o + 2, v * x.z);
    unsafeAtomicAdd(o + 3, v * x.w);
}

// ---------------------------------------------------------------------------
// Stage 4: in-place GEMM + ReLU:  out[r] = relu(agg[r] @ w_rel[r])
// Safe in place: each wave fully consumes its 16x128 input tile (data
// dependence) before writing the same tile; blocks own disjoint rows.
// (Clamped A rows may read a neighbor tile's row, but their D rows are
// never stored, and WMMA rows are independent, so no correctness impact.)
// ---------------------------------------------------------------------------
__global__ __launch_bounds__(256) void rgcn_gemm_relu(float* __restrict__ out,
                                                      const float* __restrict__ wrel,
                                                      int N) {
    __shared__ float Bs[DIM * DIM];   // 64 KB: w_rel[r], row-major [K=128][O=128]

    int r = blockIdx.y;

    // cooperative load of w_rel[r] into LDS (4096 float4, 16 per thread)
    {
        const float4* W4 = (const float4*)(wrel + (size_t)r * DIM * DIM);
        float4* B4 = (float4*)Bs;
        #pragma unroll
        for (int i = 0; i < 16; ++i)
            B4[i * 256 + threadIdx.x] = W4[i * 256 + threadIdx.x];
    }
    __syncthreads();

    int waveId  = threadIdx.x >> 5;
    int lane    = threadIdx.x & 31;
    int rowTile = blockIdx.x * 8 + waveId;
    int rowBase = rowTile * 16;
    if (rowBase >= N) return;                    // wave-uniform exit (EXEC stays full)

    int m     = lane & 15;                       // A: M index / B,D: N-within-tile
    int khalf = (lane >> 4) * 2;                 // lanes 0-15 -> K 0,1 ; 16-31 -> K 2,3

    int arow = rowBase + m;
    if (arow >= N) arow = N - 1;                 // clamp (clamped rows never stored)
    const float* Arow = out + ((size_t)r * N + arow) * DIM;

    v8f zero = {0.f, 0.f, 0.f, 0.f, 0.f, 0.f, 0.f, 0.f};
    v8f acc[8];
    #pragma unroll
    for (int t = 0; t < 8; ++t) acc[t] = zero;

    for (int kb = 0; kb < 32; ++kb) {            // K = 128, 4 per WMMA
        int kk = kb * 4 + khalf;
        v2f a;
        a.x = Arow[kk];
        a.y = Arow[kk + 1];
        #pragma unroll
        for (int t = 0; t < 8; ++t) {
            int n = t * 16 + m;
            v2f b;
            b.x = Bs[kk * DIM + n];
            b.y = Bs[(kk + 1) * DIM + n];
            acc[t] = __builtin_amdgcn_wmma_f32_16x16x4_f32(
                false, a, false, b, (short)0, acc[t], false, false);
        }
    }

    // D layout: vgpr v, lane l -> M = v + 8*(l>>4), N = (l&15) within tile
    int mHi = (lane >> 4) * 8;
    float* Obase = out + ((size_t)r * N + rowBase) * DIM + m;  // + N-offset within tile
    if (rowBase + 16 <= N) {
        // full tile: wave-uniform, no per-element guards
        #pragma unroll
        for (int t = 0; t < 8; ++t) {
            #pragma unroll
            for (int v = 0; v < 8; ++v) {
                float x = acc[t][v];
                Obase[(size_t)(mHi + v) * DIM + t * 16] = x > 0.f ? x : 0.f;
            }
        }
    } else {
        // partial tail tile: per-row guard
        #pragma unroll
        for (int t = 0; t < 8; ++t) {
            #pragma unroll
            for (int v = 0; v < 8; ++v) {
                if (rowBase + mHi + v < N) {
                    float x = acc[t][v];
                    Obase[(size_t)(mHi + v) * DIM + t * 16] = x > 0.f ? x : 0.f;
                }
            }
        }
    }
}

// ---------------------------------------------------------------------------
extern "C" void kernel_launch(void* const* d_in, const int* in_sizes, int n_in,
                              void* d_out, int out_size, void* d_ws, size_t ws_size,
                              hipStream_t stream) {
    const float* seqs   = (const float*)d_in[0];   // [1, N, 128]
    const float* comp   = (const float*)d_in[1];   // [R, NB]
    const float* weight = (const float*)d_in[2];   // [NB, 128, 128]
    const float* vals   = (const float*)d_in[3];   // [R, E]
    const int*   rows   = (const int*)d_in[4];     // [R, E]
    const int*   cols   = (const int*)d_in[5];     // [R, E]
    float* out = (float*)d_out;                    // [R, N, 128]

    int N = in_sizes[0] / DIM;
    int E = in_sizes[3] / R_REL;

    float* wrel = (float*)d_ws;                    // R*128*128 floats = 256 KB

    // 1) basis combination -> per-relation weights
    rgcn_make_wrel<<<(R_REL * DIM * DIM) / 256, 256, 0, stream>>>(comp, weight, wrel);

    // 2) zero the accumulator (== d_out)
    hipMemsetAsync(d_out, 0, (size_t)out_size * sizeof(float), stream);

    // 3) sparse aggregate of raw features: agg[r] = A_r @ seq
    {
        dim3 grid((E + 7) / 8, R_REL);
        rgcn_spmm<<<grid, 256, 0, stream>>>(seqs, vals, rows, cols, out, E, N);
    }

    // 4) in-place dense transform + ReLU: out[r] = relu(agg[r] @ w_rel[r])
    {
        int rowTiles = (N + 15) / 16;
        dim3 grid((rowTiles + 7) / 8, R_REL);
        rgcn_gemm_relu<<<grid, 256, 0, stream>>>(out, wrel, N);
    }
}